// NaiveGivensRotationLayer_40553081208879
// MI455X (gfx1250) — compile-verified
//
#include <hip/hip_runtime.h>

typedef float v2f __attribute__((ext_vector_type(2)));
typedef float v8f __attribute__((ext_vector_type(8)));

#define NN      128
#define NROT    8128      // 128*127/2
#define MAT     16384     // 128*128 floats
#define NSEG    8

// ---------------------------------------------------------------------------
// Kernel 0: precompute (cos, sin) pairs once, accurately, in parallel.
// ---------------------------------------------------------------------------
__global__ void k_cossin(const float* __restrict__ angles, float* __restrict__ cs) {
    int e = blockIdx.x * blockDim.x + threadIdx.x;
    if (e < NROT) {
        float a = angles[e];
        cs[2 * e + 0] = cosf(a);
        cs[2 * e + 1] = sinf(a);
    }
}

// ---------------------------------------------------------------------------
// Kernel 1: build 8 segment products S_p in parallel (W = S_0 * S_1 * ... * S_7).
// One workgroup per segment, one thread per row of U. U kept column-major in
// LDS (U[c*128 + t]) -> lane-consecutive, bank-conflict-free, exactly 64KB.
// Within an i-block the carried column lives in a register: 1-FMA chain/rot.
// Rotation order is the analytic triu(i<j) order, so `blocks` is not needed.
// ---------------------------------------------------------------------------
__global__ void k_segment_build(const float* __restrict__ cs, float* __restrict__ segs) {
    __shared__ float U[MAT];           // U[c*128 + t] = U_row-t_col-c, 64KB
    const int t = threadIdx.x;         // row owned by this thread, 0..127
    const int p = blockIdx.x;          // segment id, 0..NSEG-1

    // identity init: thread t touches only its own stride-128 slice
    for (int c = 0; c < NN; ++c)
        U[c * NN + t] = (c == t) ? 1.0f : 0.0f;

    // process i-blocks assigned to this segment (monotone partition by
    // cumulative rotation count L(i) = 127*i - i*(i-1)/2)
    for (int i = 0; i < NN - 1; ++i) {
        int Li = 127 * i - (i * (i - 1)) / 2;       // rotations before block i
        int s  = (Li * NSEG) / NROT;
        if (s > NSEG - 1) s = NSEG - 1;
        if (s != p) continue;

        float ci = U[i * NN + t];
        for (int j = i + 1; j < NN; ++j) {
            int   e  = Li + (j - i - 1);
            v2f   sc = *(const v2f*)(cs + 2 * e);   // uniform -> scalar load
            float c  = sc.x, sn = sc.y;
            float cj = U[j * NN + t];
            float nc = fmaf(c, ci, sn * cj);        // c*ci + s*cj
            U[j * NN + t] = fmaf(-sn, ci, c * cj);  // -s*ci + c*cj
            ci = nc;
        }
        U[i * NN + t] = ci;
    }

    // write S_p row-major: thread t owns row t; LDS reads lane-consecutive
    float* S = segs + (size_t)p * MAT;
    for (int c4 = 0; c4 < NN; c4 += 4) {
        float4 v;
        v.x = U[(c4 + 0) * NN + t];
        v.y = U[(c4 + 1) * NN + t];
        v.z = U[(c4 + 2) * NN + t];
        v.w = U[(c4 + 3) * NN + t];
        *(float4*)(S + t * NN + c4) = v;
    }
}

// ---------------------------------------------------------------------------
// Kernel 2: C = A * B for adjacent 128x128 pairs (tree-combine round).
// B is staged in LDS pre-swizzled into WMMA-B fragment order (float2/lane).
// A fragments (row-major contiguous float2) load straight from global.
// One workgroup per product, 8 waves, each wave = 16 rows of C.
// kb-outer / nt-inner: one A fragment feeds 8 WMMAs, 8 v8f accumulators live.
// ---------------------------------------------------------------------------
__global__ void k_combine(const float* __restrict__ src, float* __restrict__ dst) {
    __shared__ float bf[MAT];                     // fragment-order B, 64KB
    const float* A = src + (size_t)blockIdx.x * 2 * MAT;
    const float* B = A + MAT;
    float*       C = dst + (size_t)blockIdx.x * MAT;

    const int tid = threadIdx.x;
    for (int idx = tid; idx < 8192; idx += 256) { // 8192 float2 fragments
        int lane = idx & 31, kb = (idx >> 5) & 31, nt = idx >> 10;
        int n = nt * 16 + (lane & 15);
        int k = kb * 4 + ((lane >> 4) << 1);
        bf[2 * idx + 0] = B[(k + 0) * NN + n];    // Bmat[k][n]   = B[k][n]
        bf[2 * idx + 1] = B[(k + 1) * NN + n];    // Bmat[k+1][n]
    }
    __syncthreads();

    const int lane = tid & 31, w = tid >> 5;
    const int m = lane & 15, hk = lane >> 4;
    const int r0 = w * 16;
    const float* arow = A + (r0 + m) * NN + hk * 2;

    v8f acc[8];
#pragma unroll
    for (int nt = 0; nt < 8; ++nt)
#pragma unroll
        for (int d = 0; d < 8; ++d) acc[nt][d] = 0.0f;

#pragma unroll
    for (int kb = 0; kb < 32; ++kb) {
        v2f a = *(const v2f*)(arow + kb * 4);
#pragma unroll
        for (int nt = 0; nt < 8; ++nt) {
            v2f b = *(const v2f*)(bf + 2 * ((nt * 32 + kb) * 32 + lane));
            acc[nt] = __builtin_amdgcn_wmma_f32_16x16x4_f32(
                false, a, false, b, (short)0, acc[nt], false, false);
        }
    }

#pragma unroll
    for (int nt = 0; nt < 8; ++nt)
#pragma unroll
        for (int d = 0; d < 8; ++d)
            C[(r0 + d + 8 * hk) * NN + nt * 16 + m] = acc[nt][d];
}

// ---------------------------------------------------------------------------
// Kernel 3: out = x @ W^T + bias   (memory-bound: 512MB HBM @ 23.3TB/s ~22us)
// W staged per-block in LDS in fragment order (64KB, conflict-free ds_load_b64).
// Wave tile: 16 batch rows x 128 cols -> 256 v_wmma_f32_16x16x4_f32.
// A-fragment global loads match the 16x4 f32 WMMA A layout exactly.
// x / out are touched exactly once -> non-temporal to keep L2 for W reuse.
// ---------------------------------------------------------------------------
__global__ void __launch_bounds__(256)
k_gemm(const float* __restrict__ x, const float* __restrict__ W,
       const float* __restrict__ bias, float* __restrict__ out) {
    __shared__ float wf[MAT];                     // fragment-order W, 64KB
    const int tid = threadIdx.x;
    for (int idx = tid; idx < 8192; idx += 256) {
        int lane = idx & 31, kb = (idx >> 5) & 31, nt = idx >> 10;
        int n = nt * 16 + (lane & 15);
        int k = kb * 4 + ((lane >> 4) << 1);
        v2f v = *(const v2f*)(W + n * NN + k);    // Bmat[k][n] = W[n][k]
        *(v2f*)(wf + 2 * idx) = v;
    }

    const int lane = tid & 31, w = tid >> 5;
    const int m = lane & 15, hk = lane >> 4;
    float bv[8];
#pragma unroll
    for (int nt = 0; nt < 8; ++nt) bv[nt] = bias[nt * 16 + m];
    __syncthreads();

    const long tile = (long)blockIdx.x * 8 + w;   // 16-row batch tile per wave
    const float* xrow = x + (tile * 16 + m) * NN + hk * 2;

    v8f acc[8];
#pragma unroll
    for (int nt = 0; nt < 8; ++nt)
#pragma unroll
        for (int d = 0; d < 8; ++d) acc[nt][d] = bv[nt];

#pragma unroll
    for (int kb = 0; kb < 32; ++kb) {
        v2f a = __builtin_nontemporal_load((const v2f*)(xrow + kb * 4));
#pragma unroll
        for (int nt = 0; nt < 8; ++nt) {
            v2f b = *(const v2f*)(wf + 2 * ((nt * 32 + kb) * 32 + lane));
            acc[nt] = __builtin_amdgcn_wmma_f32_16x16x4_f32(
                false, a, false, b, (short)0, acc[nt], false, false);
        }
    }

    float* orow = out + tile * 16 * NN;
#pragma unroll
    for (int nt = 0; nt < 8; ++nt)
#pragma unroll
        for (int d = 0; d < 8; ++d)
            __builtin_nontemporal_store(acc[nt][d],
                orow + (d + 8 * hk) * NN + nt * 16 + m);
}

// ---------------------------------------------------------------------------
extern "C" void kernel_launch(void* const* d_in, const int* in_sizes, int n_in,
                              void* d_out, int out_size, void* d_ws, size_t ws_size,
                              hipStream_t stream) {
    const float* x      = (const float*)d_in[0];
    const float* angles = (const float*)d_in[1];
    const float* bias   = (const float*)d_in[2];
    // d_in[3] (blocks) unused: pair order is the analytic triu(i<j) order.
    float* out = (float*)d_out;
    float* wsf = (float*)d_ws;

    float* cs   = wsf;                 // 16256 floats (pad to 16384)
    float* buf0 = wsf + 16384;         // 8 segment matrices (8 * 16384 floats)
    float* buf1 = buf0 + NSEG * MAT;   // ping-pong (4 * 16384 floats)

    const int batch = in_sizes[0] / NN;          // 524288
    const int gemm_blocks = batch / NN;          // 8 waves * 16 rows per block

    k_cossin       <<<dim3(32), dim3(256), 0, stream>>>(angles, cs);
    k_segment_build<<<dim3(NSEG), dim3(128), 0, stream>>>(cs, buf0);
    k_combine      <<<dim3(4), dim3(256), 0, stream>>>(buf0, buf1); // 8 -> 4
    k_combine      <<<dim3(2), dim3(256), 0, stream>>>(buf1, buf0); // 4 -> 2
    k_combine      <<<dim3(1), dim3(256), 0, stream>>>(buf0, buf1); // 2 -> 1 = W
    k_gemm         <<<dim3(gemm_blocks), dim3(256), 0, stream>>>(x, buf1, bias, out);
}